// WeightedSVD_24017457119751
// MI455X (gfx1250) — compile-verified
//
#include <hip/hip_runtime.h>
#include <hip/hip_bf16.h>

typedef __attribute__((ext_vector_type(2))) float v2f;
typedef __attribute__((ext_vector_type(8))) float v8f;

// -----------------------------------------------------------------------------
// Kernel 1: per-batch streaming reduction of 16 augmented moments.
//   u = [w*x, w*y, w*z, w], v = [cx, cy, cz, 1];  M[i][j] = sum_n u_i * v_j
// One workgroup (256 thr, 8 waves) per batch; each thread owns 4 points via
// float4 loads from the contiguous N-rows. Wave partials combined with
// V_WMMA_F32_16X16X4_F32 (D = partials(16x4) * ones(4x16) + C, chained twice).
// -----------------------------------------------------------------------------
__global__ __launch_bounds__(256) void wkabsch_reduce_kernel(
    const float* __restrict__ src, const float* __restrict__ corr,
    const float* __restrict__ weight, float* __restrict__ ws, int N) {
  const int b = blockIdx.x;
  const int t = threadIdx.x;
  const int n = t * 4;  // 256 threads * 4 points = 1024 points

  const float* sb = src + (size_t)b * 3 * N;
  const float* cb = corr + (size_t)b * 3 * N;
  const float* wb = weight + (size_t)b * N;

  float4 sx = *(const float4*)(sb + 0 * N + n);
  float4 sy = *(const float4*)(sb + 1 * N + n);
  float4 sz = *(const float4*)(sb + 2 * N + n);
  float4 cx = *(const float4*)(cb + 0 * N + n);
  float4 cy = *(const float4*)(cb + 1 * N + n);
  float4 cz = *(const float4*)(cb + 2 * N + n);
  float4 w  = *(const float4*)(wb + n);

  float acc[16];
#pragma unroll
  for (int i = 0; i < 16; ++i) acc[i] = 0.0f;

#define PT(SX, SY, SZ, CX, CY, CZ, W)                                   \
  {                                                                     \
    float ux = (W) * (SX), uy = (W) * (SY), uz = (W) * (SZ);            \
    acc[0] += ux * (CX); acc[1] += ux * (CY); acc[2] += ux * (CZ);      \
    acc[3] += ux;                                                       \
    acc[4] += uy * (CX); acc[5] += uy * (CY); acc[6] += uy * (CZ);      \
    acc[7] += uy;                                                       \
    acc[8] += uz * (CX); acc[9] += uz * (CY); acc[10] += uz * (CZ);     \
    acc[11] += uz;                                                      \
    acc[12] += (W) * (CX); acc[13] += (W) * (CY); acc[14] += (W) * (CZ);\
    acc[15] += (W);                                                     \
  }
  PT(sx.x, sy.x, sz.x, cx.x, cy.x, cz.x, w.x)
  PT(sx.y, sy.y, sz.y, cx.y, cy.y, cz.y, w.y)
  PT(sx.z, sy.z, sz.z, cx.z, cy.z, cz.z, w.z)
  PT(sx.w, sy.w, sz.w, cx.w, cy.w, cz.w, w.w)
#undef PT

  // Intra-wave butterfly reduction (wave32).
#pragma unroll
  for (int i = 0; i < 16; ++i) {
    float v = acc[i];
    v += __shfl_xor(v, 16, 32);
    v += __shfl_xor(v, 8, 32);
    v += __shfl_xor(v, 4, 32);
    v += __shfl_xor(v, 2, 32);
    v += __shfl_xor(v, 1, 32);
    acc[i] = v;
  }

  __shared__ float lds[8 * 16];  // [wave][moment]
  const int lane = t & 31;
  const int wave = t >> 5;
  if (lane == 0) {
#pragma unroll
    for (int i = 0; i < 16; ++i) lds[wave * 16 + i] = acc[i];
  }
  __syncthreads();

  // Wave 0: combine 8 wave-partials with two chained f32 WMMAs.
  // A layout (16x4 f32): lanes 0-15 hold A[m][0..1], lanes 16-31 hold A[m][2..3]
  // with A[m][k] = lds[k*16 + m]; B = ones(4x16) -> D[m][*] = sum_k A[m][k].
  if (t < 32) {
    const int m = lane & 15;
    const int kb = (lane < 16) ? 0 : 2;
    v2f a0, a1, bf;
    a0.x = lds[(kb + 0) * 16 + m];
    a0.y = lds[(kb + 1) * 16 + m];
    a1.x = lds[(kb + 4) * 16 + m];
    a1.y = lds[(kb + 5) * 16 + m];
    bf.x = 1.0f;
    bf.y = 1.0f;
    v8f c = {};
    c = __builtin_amdgcn_wmma_f32_16x16x4_f32(false, a0, false, bf, (short)0,
                                              c, false, false);
    c = __builtin_amdgcn_wmma_f32_16x16x4_f32(false, a1, false, bf, (short)0,
                                              c, false, false);
    // D layout: VGPR r -> lanes 0-15 give M=r, lanes 16-31 give M=8+r.
    float* wsb = ws + (size_t)b * 16;
    if (lane == 0) {
#pragma unroll
      for (int r = 0; r < 8; ++r) wsb[r] = c[r];
    }
    if (lane == 16) {
#pragma unroll
      for (int r = 0; r < 8; ++r) wsb[8 + r] = c[r];
    }
  }
}

// -----------------------------------------------------------------------------
// Kernel 2: per-batch Horn quaternion solve (equivalent to det-corrected
// Kabsch). Fully unrolled 4x4 Jacobi eigensolver, registers only.
// -----------------------------------------------------------------------------
template <int P, int Q>
__device__ inline void jacobi_rot(float K[4][4], float V[4][4]) {
  float apq = K[P][Q];
  if (__builtin_fabsf(apq) < 1e-30f) return;
  float app = K[P][P], aqq = K[Q][Q];
  float tau = (aqq - app) / (2.0f * apq);
  float tt = __builtin_copysignf(1.0f, tau) /
             (__builtin_fabsf(tau) + sqrtf(1.0f + tau * tau));
  float cc = rsqrtf(1.0f + tt * tt);
  float ss = tt * cc;
#pragma unroll
  for (int k = 0; k < 4; ++k) {  // K = K * J
    float kp = K[k][P], kq = K[k][Q];
    K[k][P] = cc * kp - ss * kq;
    K[k][Q] = ss * kp + cc * kq;
  }
#pragma unroll
  for (int k = 0; k < 4; ++k) {  // K = J^T * K
    float kp = K[P][k], kq = K[Q][k];
    K[P][k] = cc * kp - ss * kq;
    K[Q][k] = ss * kp + cc * kq;
  }
#pragma unroll
  for (int k = 0; k < 4; ++k) {  // V = V * J
    float vp = V[k][P], vq = V[k][Q];
    V[k][P] = cc * vp - ss * vq;
    V[k][Q] = ss * vp + cc * vq;
  }
}

__global__ __launch_bounds__(256) void wkabsch_solve_kernel(
    const float* __restrict__ ws, float* __restrict__ outR,
    float* __restrict__ outT, int B) {
  const int b = blockIdx.x * blockDim.x + threadIdx.x;
  if (b >= B) return;
  const float* s = ws + (size_t)b * 16;

  const float S1 = s[15];
  const float inv = 1.0f / S1;
  float Sx[3], Sy[3];
#pragma unroll
  for (int i = 0; i < 3; ++i) Sx[i] = s[i * 4 + 3];
#pragma unroll
  for (int j = 0; j < 3; ++j) Sy[j] = s[12 + j];

  float H[3][3];
#pragma unroll
  for (int i = 0; i < 3; ++i)
#pragma unroll
    for (int j = 0; j < 3; ++j) H[i][j] = s[i * 4 + j] - Sx[i] * Sy[j] * inv;

  float cx0 = Sx[0] * inv, cx1 = Sx[1] * inv, cx2 = Sx[2] * inv;
  float cy0 = Sy[0] * inv, cy1 = Sy[1] * inv, cy2 = Sy[2] * inv;

  // Besl-McKay symmetric 4x4 from cross-covariance H (H_ij = sum w a_i b_j).
  const float tr = H[0][0] + H[1][1] + H[2][2];
  float K[4][4];
  K[0][0] = tr;
  K[0][1] = K[1][0] = H[1][2] - H[2][1];
  K[0][2] = K[2][0] = H[2][0] - H[0][2];
  K[0][3] = K[3][0] = H[0][1] - H[1][0];
#pragma unroll
  for (int i = 0; i < 3; ++i)
#pragma unroll
    for (int j = 0; j < 3; ++j)
      K[1 + i][1 + j] = H[i][j] + H[j][i] - ((i == j) ? tr : 0.0f);

  float V[4][4];
#pragma unroll
  for (int i = 0; i < 4; ++i)
#pragma unroll
    for (int j = 0; j < 4; ++j) V[i][j] = (i == j) ? 1.0f : 0.0f;

#pragma unroll
  for (int sweep = 0; sweep < 8; ++sweep) {
    jacobi_rot<0, 1>(K, V);
    jacobi_rot<0, 2>(K, V);
    jacobi_rot<0, 3>(K, V);
    jacobi_rot<1, 2>(K, V);
    jacobi_rot<1, 3>(K, V);
    jacobi_rot<2, 3>(K, V);
  }

  // Max-eigenvalue column -> quaternion (scalar-first).
  float d0 = K[0][0], d1 = K[1][1], d2 = K[2][2], d3 = K[3][3];
  float q0, q1, q2, q3;
  {
    float bv = d0;
    q0 = V[0][0]; q1 = V[1][0]; q2 = V[2][0]; q3 = V[3][0];
    if (d1 > bv) { bv = d1; q0 = V[0][1]; q1 = V[1][1]; q2 = V[2][1]; q3 = V[3][1]; }
    if (d2 > bv) { bv = d2; q0 = V[0][2]; q1 = V[1][2]; q2 = V[2][2]; q3 = V[3][2]; }
    if (d3 > bv) {          q0 = V[0][3]; q1 = V[1][3]; q2 = V[2][3]; q3 = V[3][3]; }
  }
  float nrm = q0 * q0 + q1 * q1 + q2 * q2 + q3 * q3;
  if (nrm < 1e-30f) { q0 = 1.0f; q1 = q2 = q3 = 0.0f; nrm = 1.0f; }
  const float qi = rsqrtf(nrm);
  q0 *= qi; q1 *= qi; q2 *= qi; q3 *= qi;

  float R[3][3];
  R[0][0] = 1.0f - 2.0f * (q2 * q2 + q3 * q3);
  R[0][1] = 2.0f * (q1 * q2 - q0 * q3);
  R[0][2] = 2.0f * (q1 * q3 + q0 * q2);
  R[1][0] = 2.0f * (q1 * q2 + q0 * q3);
  R[1][1] = 1.0f - 2.0f * (q1 * q1 + q3 * q3);
  R[1][2] = 2.0f * (q2 * q3 - q0 * q1);
  R[2][0] = 2.0f * (q1 * q3 - q0 * q2);
  R[2][1] = 2.0f * (q2 * q3 + q0 * q1);
  R[2][2] = 1.0f - 2.0f * (q1 * q1 + q2 * q2);

  float* Rb = outR + (size_t)b * 9;
#pragma unroll
  for (int i = 0; i < 3; ++i)
#pragma unroll
    for (int j = 0; j < 3; ++j) Rb[i * 3 + j] = R[i][j];

  float* Tb = outT + (size_t)b * 3;
  Tb[0] = cy0 - (R[0][0] * cx0 + R[0][1] * cx1 + R[0][2] * cx2);
  Tb[1] = cy1 - (R[1][0] * cx0 + R[1][1] * cx1 + R[1][2] * cx2);
  Tb[2] = cy2 - (R[2][0] * cx0 + R[2][1] * cx1 + R[2][2] * cx2);
}

extern "C" void kernel_launch(void* const* d_in, const int* in_sizes, int n_in,
                              void* d_out, int out_size, void* d_ws,
                              size_t ws_size, hipStream_t stream) {
  const float* src = (const float*)d_in[0];
  const float* corr = (const float*)d_in[1];
  const float* weight = (const float*)d_in[2];
  const int N = 1024;
  const int B = in_sizes[0] / (3 * N);  // 4096

  float* ws = (float*)d_ws;             // B * 16 floats = 256 KB
  float* out = (float*)d_out;
  float* outR = out;                    // B*9 floats
  float* outT = out + (size_t)B * 9;    // B*3 floats

  wkabsch_reduce_kernel<<<B, 256, 0, stream>>>(src, corr, weight, ws, N);
  wkabsch_solve_kernel<<<(B + 255) / 256, 256, 0, stream>>>(ws, outR, outT, B);
}